// EncodingLoss_36601711297183
// MI455X (gfx1250) — compile-verified
//
#include <hip/hip_runtime.h>

#define CL   10
#define NAc  2048
#define NRc  2048
#define DMc  64
#define NPc  4096
#define POSCAP (NAc * 512)

typedef __attribute__((ext_vector_type(2))) float v2f;
typedef __attribute__((ext_vector_type(8))) float v8f;

// ---------------- helpers ----------------
__device__ __forceinline__ float warpSum(float v) {
#pragma unroll
  for (int m = 16; m; m >>= 1) v += __shfl_xor(v, m, 32);
  return v;
}
__device__ __forceinline__ float blockSum256(float v, float* red) {
  int tid = threadIdx.x;
  red[tid] = v; __syncthreads();
  for (int s = 128; s; s >>= 1) { if (tid < s) red[tid] += red[tid + s]; __syncthreads(); }
  float r = red[0]; __syncthreads();
  return r;
}
__device__ __forceinline__ int blockSum256i(int v, int* red) {
  int tid = threadIdx.x;
  red[tid] = v; __syncthreads();
  for (int s = 128; s; s >>= 1) { if (tid < s) red[tid] += red[tid + s]; __syncthreads(); }
  int r = red[0]; __syncthreads();
  return r;
}
// order-preserving float<->uint key (for atomic min on floats)
__device__ __forceinline__ unsigned fkey(float f) {
  unsigned b = __float_as_uint(f);
  return (b & 0x80000000u) ? ~b : (b | 0x80000000u);
}
__device__ __forceinline__ float fdec(unsigned k) {
  return (k & 0x80000000u) ? __uint_as_float(k ^ 0x80000000u) : __uint_as_float(~k);
}

// ---------------- init ----------------
__global__ void init_kernel(int* counts, int* ivars, float* scal, unsigned* minBits,
                            float* rnaMeanSum, float* psum, float* s_c) {
  int t = threadIdx.x;
  if (t < CL) counts[t] = 0;
  if (t < 8)  ivars[t] = 0;
  if (t < 32) scal[t] = 0.0f;
  if (t < CL) minBits[t] = 0xFFFFFFFFu;
  if (t < CL * DMc) rnaMeanSum[t] = 0.0f;
  if (t < CL * CL) psum[t] = 0.0f;
  if (t < CL) s_c[t] = 0.0f;
}

__global__ void counts_kernel(const int* __restrict__ lab, int* counts) {
  int n = blockIdx.x * 256 + threadIdx.x;
  if (n < NRc) atomicAdd(&counts[lab[n]], 1);
}

__global__ void kmin_kernel(const int* counts, int* ivars) {
  int lane = threadIdx.x & 31;
  int v = (lane < CL) ? counts[lane] : 0x7FFFFFFF;
#pragma unroll
  for (int m = 16; m; m >>= 1) { int o = __shfl_xor(v, m, 32); v = (o < v) ? o : v; }
  if (lane == 0) ivars[0] = v + 1;
}

// ---------------- row normalization ----------------
__global__ void rownorm_rna_kernel(const float* __restrict__ Er, const int* __restrict__ lab,
                                   float* __restrict__ Ern, float* rnaMeanSum) {
  int wave = threadIdx.x >> 5, lane = threadIdx.x & 31;
  int n = blockIdx.x * 8 + wave;
  float a = Er[n * DMc + lane];
  float b = Er[n * DMc + lane + 32];
  float ss = warpSum(a * a + b * b);
  float inv = rsqrtf(ss);
  Ern[n * DMc + lane] = a * inv;
  Ern[n * DMc + lane + 32] = b * inv;
  int c = lab[n];
  atomicAdd(&rnaMeanSum[c * DMc + lane], a);
  atomicAdd(&rnaMeanSum[c * DMc + lane + 32], b);
}

__global__ void rownorm_atac_kernel(const float* __restrict__ Ea, float* __restrict__ Ean) {
  int wave = threadIdx.x >> 5, lane = threadIdx.x & 31;
  int n = blockIdx.x * 8 + wave;
  float a = Ea[n * DMc + lane];
  float b = Ea[n * DMc + lane + 32];
  float inv = rsqrtf(warpSum(a * a + b * b));
  Ean[n * DMc + lane] = a * inv;
  Ean[n * DMc + lane + 32] = b * inv;
}

__global__ void peak_norm_kernel(const float* __restrict__ P, float* __restrict__ peakInv) {
  __shared__ float red[256];
  int i = blockIdx.x;
  float s = 0.f;
  for (int j = threadIdx.x; j < NPc; j += 256) { float v = P[(size_t)i * NPc + j]; s += v * v; }
  s = blockSum256(s, red);
  if (threadIdx.x == 0) peakInv[i] = rsqrtf(s);
}

// ---------------- column stats ----------------
__global__ void colstats_kernel(const float* __restrict__ Er, const float* __restrict__ Ea,
                                float* colMeanR, float* colMeanA, float* colSqA, float* scal) {
  __shared__ float red[256];
  int which = blockIdx.x / DMc;
  int col = blockIdx.x % DMc;
  const float* E = which ? Ea : Er;
  float s = 0.f, sq = 0.f, sab = 0.f;
  for (int n = threadIdx.x; n < NRc; n += 256) {
    float v = E[n * DMc + col];
    s += v; sq += v * v; sab += fabsf(v);
  }
  s = blockSum256(s, red);
  sq = blockSum256(sq, red);
  sab = blockSum256(sab, red);
  if (threadIdx.x == 0) {
    if (which == 0) colMeanR[col] = s / (float)NRc;
    else { colMeanA[col] = s / (float)NAc; colSqA[col] = sq; atomicAdd(&scal[2], sab); }
  }
}

__global__ void cov_abs_kernel(const float* __restrict__ Er, const float* __restrict__ Ea,
                               const float* colMeanR, const float* colMeanA, float* scal) {
  __shared__ float red[256];
  int gid = blockIdx.x * 256 + threadIdx.x;
  int which = gid >> 12;          // blocks 0..15 -> Er, 16..31 -> Ea (uniform per block)
  int pair = gid & 4095;
  int c1 = pair >> 6, c2 = pair & 63;
  const float* E = which ? Ea : Er;
  const float* mu = which ? colMeanA : colMeanR;
  float m1 = mu[c1], m2 = mu[c2], acc = 0.f;
  for (int n = 0; n < NRc; ++n)
    acc += (E[n * DMc + c1] - m1) * (E[n * DMc + c2] - m2);
  float v = fabsf(acc / (float)(NRc - 1));
  v = blockSum256(v, red);
  if (threadIdx.x == 0) atomicAdd(&scal[which ? 1 : 0], v);
}

// ---------------- WMMA X @ Y^T, 32x32 register block per wave (2x2 tiles) ----------------
__device__ __forceinline__ void wmma_store_tile(float* __restrict__ Out, v8f acc,
                                                int rowBase, int col, int Nt,
                                                const float* __restrict__ invX,
                                                float sInvCol, int zeroDiag) {
#pragma unroll
  for (int r = 0; r < 8; ++r) {
    int row = rowBase + r;
    float v = acc[r];
    if (invX) v *= invX[row];
    v *= sInvCol;
    if (zeroDiag && row == col) v = 0.0f;
    Out[(size_t)row * Nt + col] = v;
  }
}

__global__ void wmma_xyT_kernel(const float* __restrict__ X, const float* __restrict__ Y,
                                float* __restrict__ Out, int Mt, int Nt, int K,
                                const float* __restrict__ invX, const float* __restrict__ invY,
                                int zeroDiag) {
  int wave = (blockIdx.x * blockDim.x + threadIdx.x) >> 5;
  int lane = threadIdx.x & 31;
  int tilesN = Nt >> 5;                      // 32-wide blocks
  int totalTiles = (Mt >> 5) * tilesN;
  if (wave >= totalTiles) return;
  int ti = wave / tilesN, tj = wave % tilesN;
  int koff = (lane >> 4) * 2;
  int mrow0 = ti * 32 + (lane & 15);
  int ncol0 = tj * 32 + (lane & 15);
  const float* x0 = X + (size_t)mrow0 * K + koff;
  const float* x1 = x0 + (size_t)16 * K;
  const float* y0 = Y + (size_t)ncol0 * K + koff;
  const float* y1 = y0 + (size_t)16 * K;
  v8f acc00 = {}, acc01 = {}, acc10 = {}, acc11 = {};
  for (int kk = 0; kk < K; kk += 4) {
    v2f a0 = *(const v2f*)(x0 + kk);
    v2f a1 = *(const v2f*)(x1 + kk);
    v2f b0 = *(const v2f*)(y0 + kk);
    v2f b1 = *(const v2f*)(y1 + kk);
    acc00 = __builtin_amdgcn_wmma_f32_16x16x4_f32(false, a0, false, b0, (short)0, acc00, false, false);
    acc01 = __builtin_amdgcn_wmma_f32_16x16x4_f32(false, a0, false, b1, (short)0, acc01, false, false);
    acc10 = __builtin_amdgcn_wmma_f32_16x16x4_f32(false, a1, false, b0, (short)0, acc10, false, false);
    acc11 = __builtin_amdgcn_wmma_f32_16x16x4_f32(false, a1, false, b1, (short)0, acc11, false, false);
  }
  int rsub = (lane >> 4) << 3;               // 0 or 8
  int csub = lane & 15;
  int r0 = ti * 32 + rsub, r1 = r0 + 16;
  int c0 = tj * 32 + csub, c1 = c0 + 16;
  float si0 = invY ? invY[c0] : 1.0f;
  float si1 = invY ? invY[c1] : 1.0f;
  wmma_store_tile(Out, acc00, r0, c0, Nt, invX, si0, zeroDiag);
  wmma_store_tile(Out, acc01, r0, c1, Nt, invX, si1, zeroDiag);
  wmma_store_tile(Out, acc10, r1, c0, Nt, invX, si0, zeroDiag);
  wmma_store_tile(Out, acc11, r1, c1, Nt, invX, si1, zeroDiag);
}

// ---------------- cluster block sums of S ----------------
__global__ void sc_reduce_kernel(const float* __restrict__ S, const int* __restrict__ lab,
                                 float* s_c) {
  __shared__ float bins[CL];
  int tid = threadIdx.x;
  if (tid < CL) bins[tid] = 0.f;
  __syncthreads();
  size_t idx = (size_t)blockIdx.x * 256 + tid;
  int i = (int)(idx >> 11), j = (int)(idx & 2047);
  int li = lab[i], lj = lab[j];
  if (li == lj) atomicAdd(&bins[li], S[idx]);
  __syncthreads();
  if (tid < CL) atomicAdd(&s_c[tid], bins[tid]);
}

// ---------------- pairwise L1 cluster sums ----------------
__global__ void dl1_psum_kernel(const float* __restrict__ Er, const int* __restrict__ lab,
                                float* psum) {
  __shared__ float bins[CL * CL];
  int tid = threadIdx.x;
  if (tid < CL * CL) bins[tid] = 0.f;
  __syncthreads();
  int ti = blockIdx.x / (NRc / 16), tj = blockIdx.x % (NRc / 16);
  int i = ti * 16 + (tid >> 4);
  int j = tj * 16 + (tid & 15);
  float s = 0.f;
#pragma unroll 8
  for (int d = 0; d < DMc; ++d) s += fabsf(Er[i * DMc + d] - Er[j * DMc + d]);
  atomicAdd(&bins[lab[i] * CL + lab[j]], s);
  __syncthreads();
  if (tid < CL * CL) atomicAdd(&psum[tid], bins[tid]);
}

// ---------------- rna cluster means / alignment ----------------
__global__ void rna_mean_norm_kernel(const float* rnaMeanSum, const int* counts,
                                     float* rnaMean, float* Mn) {
  __shared__ float red[64];
  int c = blockIdx.x, d = threadIdx.x;
  float cf = (float)counts[c];
  float m = rnaMeanSum[c * DMc + d] / cf;
  rnaMean[c * DMc + d] = m;
  red[d] = m * m; __syncthreads();
  for (int s = 32; s; s >>= 1) { if (d < s) red[d] += red[d + s]; __syncthreads(); }
  Mn[c * DMc + d] = m * rsqrtf(red[0]);
}

__global__ void rna_min_sim_kernel(const float* __restrict__ Ern, const int* __restrict__ lab,
                                   const float* __restrict__ Mn, unsigned* minBits) {
  int wave = threadIdx.x >> 5, lane = threadIdx.x & 31;
  int n = blockIdx.x * 8 + wave;
  int c = lab[n];
  float p = Ern[n * DMc + lane] * Mn[c * DMc + lane]
          + Ern[n * DMc + lane + 32] * Mn[c * DMc + lane + 32];
  p = warpSum(p);
  if (lane == 0) atomicMin(&minBits[c], fkey(p));
}

__global__ void thr2_kernel(const unsigned* minBits, float* scal) {
  int lane = threadIdx.x & 31;
  float v = (lane < CL) ? fdec(minBits[lane]) : 0.f;
  v = warpSum(v);
  if (lane == 0) scal[6] = v / (float)CL;
}

__global__ void align_l1_kernel(const float* __restrict__ Ean, const float* __restrict__ Ea,
                                const float* __restrict__ Mn, const float* __restrict__ rnaMean,
                                const float* scal, float* acc) {
  int wave = threadIdx.x >> 5, lane = threadIdx.x & 31;
  int n = blockIdx.x * 8 + wave;
  float thr2 = scal[6];
  float e0 = Ean[n * DMc + lane], e1 = Ean[n * DMc + lane + 32];
  float best = -3.4e38f; int bestc = 0;
  for (int c = 0; c < CL; ++c) {
    float p = e0 * Mn[c * DMc + lane] + e1 * Mn[c * DMc + lane + 32];
    p = warpSum(p);
    if (p > best) { best = p; bestc = c; }
  }
  if (best > thr2) {
    float s = fabsf(Ea[n * DMc + lane] - rnaMean[bestc * DMc + lane])
            + fabsf(Ea[n * DMc + lane + 32] - rnaMean[bestc * DMc + lane + 32]);
    s = warpSum(s);
    if (lane == 0) atomicAdd(acc, s / (float)DMc);
  }
}

// ---------------- top-k rank mask (g), row threshold, positive compaction ----------------
__global__ void topk_kernel(const float* __restrict__ A, float* __restrict__ G,
                            float* __restrict__ rowThr, const int* ivars,
                            float* posvals, int* nposPtr) {
  __shared__ float row[NAc];
  __shared__ float red[256];
  int i = blockIdx.x, tid = threadIdx.x;
  int k = ivars[0];
  const float* arow = A + (size_t)i * NAc;
  for (int j = tid; j < NAc; j += 256) row[j] = arow[j];
  __syncthreads();
  float gsum = 0.f;
  for (int j = tid; j < NAc; j += 256) {
    float v = row[j];
    int cnt = 0;
    for (int l = 0; l < NAc; ++l) {
      float u = row[l];
      cnt += (u > v) || (u == v && l < j);
      if (cnt >= k) break;
    }
    float g = (cnt < k) ? v : 0.f;
    G[(size_t)i * NAc + j] = g;
    gsum += g;
    if (cnt < k && v > 0.f) {
      int idx = atomicAdd(nposPtr, 1);
      if (idx < POSCAP) posvals[idx] = v;
    }
  }
  gsum = blockSum256(gsum, red);
  if (tid == 0) rowThr[i] = gsum / (float)k;
}

// ---------------- global 10% quantile of positives (binary search) ----------------
__global__ void quantile_kernel(const float* __restrict__ posvals, const int* ivars, float* scal) {
  __shared__ int red[1024];
  int tid = threadIdx.x;
  int npos = ivars[1];
  if (npos > POSCAP) npos = POSCAP;
  if (npos <= 0) { if (tid == 0) scal[7] = 0.f; return; }
  float posf = 0.1f * (float)(npos - 1);
  int lo = (int)floorf(posf);
  int hi = (int)ceilf(posf);
  int targ[2] = { lo, hi };
  float sv[2];
  for (int t = 0; t < 2; ++t) {
    float a = 0.f, b = 1.5f;
    for (int it = 0; it < 40; ++it) {
      float mid = 0.5f * (a + b);
      int cnt = 0;
      for (int idx = tid; idx < npos; idx += 1024) cnt += (posvals[idx] <= mid);
      red[tid] = cnt; __syncthreads();
      for (int s = 512; s; s >>= 1) { if (tid < s) red[tid] += red[tid + s]; __syncthreads(); }
      int total = red[0]; __syncthreads();
      if (total >= targ[t] + 1) b = mid; else a = mid;
    }
    sv[t] = b;
  }
  if (tid == 0) scal[7] = sv[0] + (posf - (float)lo) * (sv[1] - sv[0]);
}

// ---------------- neighbourhood loss ----------------
__global__ void near_kernel(const float* __restrict__ G, const float* __restrict__ AC,
                            const float* __restrict__ LOW, const float* __restrict__ rowThr,
                            const float* scal, float* acc) {
  __shared__ float redF[256];
  __shared__ int redI[256];
  int i = blockIdx.x, tid = threadIdx.x;
  float q = scal[7], rt = rowThr[i];
  float spos = 0.f, sneg = 0.f; int npr = 0;
  for (int j = tid; j < NAc; j += 256) {
    size_t idx = (size_t)i * NAc + j;
    float gg = G[idx];
    if (gg < rt || gg < q) gg = 0.f;
    float lw = LOW[idx];
    if (gg > 0.f) { spos += gg * lw; npr++; }
    else sneg += (1.f - AC[idx]) * 0.5f * lw;
  }
  spos = blockSum256(spos, redF);
  sneg = blockSum256(sneg, redF);
  npr = blockSum256i(npr, redI);
  if (tid == 0) {
    float sii = spos / (float)(npr + 1);
    float sij = sneg / (float)(NAc - npr);
    atomicAdd(acc, -sii + 0.1f * sij);
  }
}

// ---------------- mutual nearest neighbour loss ----------------
__global__ void mnn_kernel(const float* __restrict__ AR, const float* __restrict__ Ea,
                           const float* __restrict__ Er, const int* ivars, float* acc) {
  __shared__ float bval[256];
  __shared__ int bidx[256];
  __shared__ int redI[256];
  __shared__ float redF[256];
  int i = blockIdx.x, tid = threadIdx.x;
  int k = ivars[0];
  const float* row = AR + (size_t)i * NRc;
  float bv = -3.4e38f; int bi = NRc;
  for (int j = tid; j < NRc; j += 256) {
    float v = row[j];
    if (v > bv || (v == bv && j < bi)) { bv = v; bi = j; }
  }
  bval[tid] = bv; bidx[tid] = bi; __syncthreads();
  for (int s = 128; s; s >>= 1) {
    if (tid < s) {
      float v = bval[tid + s]; int x = bidx[tid + s];
      if (v > bval[tid] || (v == bval[tid] && x < bidx[tid])) { bval[tid] = v; bidx[tid] = x; }
    }
    __syncthreads();
  }
  int jstar = bidx[0]; float vstar = bval[0];
  __syncthreads();
  int cnt = 0;
  for (int a = tid; a < NAc; a += 256) {
    float u = AR[(size_t)a * NRc + jstar];
    cnt += (u > vstar) || (u == vstar && a < i);
  }
  int rank = blockSum256i(cnt, redI);
  if (rank < k) {
    float s = 0.f;
    for (int d = tid; d < DMc; d += 256) s += fabsf(Ea[i * DMc + d] - Er[jstar * DMc + d]);
    s = blockSum256(s, redF);
    if (tid == 0) atomicAdd(acc, s / (float)DMc);
  }
}

// ---------------- finalize ----------------
__global__ void finalize_kernel(const int* counts, const float* psum, const float* s_c,
                                const float* colMeanA, const float* colSqA,
                                const float* scal, float* out) {
  if (threadIdx.x != 0 || blockIdx.x != 0) return;
  float cf[CL];
  for (int c = 0; c < CL; ++c) cf[c] = (float)counts[c];
  float dm[CL * CL]; float tsum = 0.f;
  for (int a = 0; a < CL; ++a)
    for (int b = 0; b < CL; ++b) {
      float v = (a == b) ? 0.f : psum[a * CL + b] / (cf[a] * cf[b] * (float)DMc);
      dm[a * CL + b] = v; tsum += v;
    }
  float thr = tsum / (float)(CL * CL);
  float ssum = 0.f; int scnt = 0;
  for (int x = 0; x < CL * CL; ++x) if (dm[x] < thr) { ssum += dm[x]; scnt++; }
  float rna_cluster = 1.f / (ssum / (float)scnt);
  float common = 0.f;
  for (int c = 0; c < CL; ++c)
    if (counts[c] > 1) common += -((s_c[c] - cf[c]) / (2.f * cf[c] * cf[c]));
  common /= (float)CL;
  float rna_red = rna_cluster + scal[0] / (float)(DMc * DMc) + common;

  float meanstd = 0.f;
  for (int d = 0; d < DMc; ++d) {
    float m = colMeanA[d];
    float var = (colSqA[d] - (float)NAc * m * m) / (float)(NAc - 1);
    meanstd += sqrtf(var);
  }
  meanstd /= (float)DMc;
  float atac_red = 1.f / meanstd + scal[1] / (float)(DMc * DMc)
                 + scal[2] / ((float)NAc * (float)DMc);

  float near_loss = scal[3] / (float)NAc;
  float mnn_loss  = scal[4] / (float)NAc;
  float L1_loss   = scal[5] / (float)NAc;

  float raresum = 0.f; int rarecnt = 0;
  for (int c = 0; c < CL; ++c)
    if ((float)counts[c] < 0.03f * (float)NRc) { raresum += (float)counts[c]; rarecnt++; }
  float L_RD = 0.25f + raresum / (float)NRc;
  float L1w = 1.f - (float)rarecnt / (float)CL;

  out[0] = L_RD * (rna_red + atac_red + near_loss) + mnn_loss + L1w * L1_loss;
}

// ---------------- host ----------------
extern "C" void kernel_launch(void* const* d_in, const int* in_sizes, int n_in,
                              void* d_out, int out_size, void* d_ws, size_t ws_size,
                              hipStream_t stream) {
  (void)in_sizes; (void)n_in; (void)out_size; (void)ws_size;
  const float* Ea   = (const float*)d_in[0];
  const float* Er   = (const float*)d_in[1];
  const int*   lab  = (const int*)d_in[2];
  const float* peak = (const float*)d_in[3];
  float* out = (float*)d_out;

  char* base = (char*)d_ws;
  size_t off = 0;
  auto take = [&](size_t bytes) -> void* {
    void* p = base + off;
    off = (off + bytes + 255) & ~(size_t)255;
    return p;
  };
  int*      counts     = (int*)take(CL * 4);
  int*      ivars      = (int*)take(8 * 4);
  float*    scal       = (float*)take(32 * 4);
  unsigned* minBits    = (unsigned*)take(CL * 4);
  float*    rnaMeanSum = (float*)take(CL * DMc * 4);
  float*    rnaMean    = (float*)take(CL * DMc * 4);
  float*    Mn         = (float*)take(CL * DMc * 4);
  float*    colMeanR   = (float*)take(DMc * 4);
  float*    colMeanA   = (float*)take(DMc * 4);
  float*    colSqA     = (float*)take(DMc * 4);
  float*    psum       = (float*)take(CL * CL * 4);
  float*    s_c        = (float*)take(CL * 4);
  float*    rowThr     = (float*)take(NAc * 4);
  float*    peakInv    = (float*)take(NAc * 4);
  float*    Ern        = (float*)take((size_t)NRc * DMc * 4);
  float*    Ean        = (float*)take((size_t)NAc * DMc * 4);
  float*    posvals    = (float*)take((size_t)POSCAP * 4);
  float*    B_Sg       = (float*)take((size_t)NAc * NAc * 4);  // S, then reused as g
  float*    B_cos      = (float*)take((size_t)NAc * NAc * 4);
  float*    B_low      = (float*)take((size_t)NAc * NAc * 4);
  float*    B_ar       = (float*)take((size_t)NAc * NAc * 4);

  init_kernel<<<1, 1024, 0, stream>>>(counts, ivars, scal, minBits, rnaMeanSum, psum, s_c);
  counts_kernel<<<NRc / 256, 256, 0, stream>>>(lab, counts);
  kmin_kernel<<<1, 32, 0, stream>>>(counts, ivars);

  rownorm_rna_kernel<<<NRc / 8, 256, 0, stream>>>(Er, lab, Ern, rnaMeanSum);
  rownorm_atac_kernel<<<NAc / 8, 256, 0, stream>>>(Ea, Ean);
  peak_norm_kernel<<<NAc, 256, 0, stream>>>(peak, peakInv);

  colstats_kernel<<<2 * DMc, 256, 0, stream>>>(Er, Ea, colMeanR, colMeanA, colSqA, scal);
  cov_abs_kernel<<<32, 256, 0, stream>>>(Er, Ea, colMeanR, colMeanA, scal);

  // 32x32 output block per wave -> (2048/32)^2 = 4096 waves -> 1024 blocks of 4 waves
  const int wblocks = ((NAc / 32) * (NAc / 32)) / 4;
  // S = Ern @ Ern^T (diag kept = 1)
  wmma_xyT_kernel<<<wblocks, 128, 0, stream>>>(Ern, Ern, B_Sg, NRc, NRc, DMc,
                                               nullptr, nullptr, 0);
  sc_reduce_kernel<<<(NRc * NRc) / 256, 256, 0, stream>>>(B_Sg, lab, s_c);
  dl1_psum_kernel<<<(NRc / 16) * (NRc / 16), 256, 0, stream>>>(Er, lab, psum);

  // atac_cos = (peak @ peak^T) scaled by row inverse norms, diag zeroed
  wmma_xyT_kernel<<<wblocks, 128, 0, stream>>>(peak, peak, B_cos, NAc, NAc, NPc,
                                               peakInv, peakInv, 1);
  // low = Ean @ Ean^T, diag zeroed
  wmma_xyT_kernel<<<wblocks, 128, 0, stream>>>(Ean, Ean, B_low, NAc, NAc, DMc,
                                               nullptr, nullptr, 1);
  // atac_rna = Ean @ Ern^T
  wmma_xyT_kernel<<<wblocks, 128, 0, stream>>>(Ean, Ern, B_ar, NAc, NRc, DMc,
                                               nullptr, nullptr, 0);

  rna_mean_norm_kernel<<<CL, 64, 0, stream>>>(rnaMeanSum, counts, rnaMean, Mn);
  rna_min_sim_kernel<<<NRc / 8, 256, 0, stream>>>(Ern, lab, Mn, minBits);
  thr2_kernel<<<1, 32, 0, stream>>>(minBits, scal);
  align_l1_kernel<<<NAc / 8, 256, 0, stream>>>(Ean, Ea, Mn, rnaMean, scal, &scal[5]);

  topk_kernel<<<NAc, 256, 0, stream>>>(B_cos, B_Sg, rowThr, ivars, posvals, &ivars[1]);
  quantile_kernel<<<1, 1024, 0, stream>>>(posvals, ivars, scal);
  near_kernel<<<NAc, 256, 0, stream>>>(B_Sg, B_cos, B_low, rowThr, scal, &scal[3]);
  mnn_kernel<<<NAc, 256, 0, stream>>>(B_ar, Ea, Er, ivars, &scal[4]);

  finalize_kernel<<<1, 1, 0, stream>>>(counts, psum, s_c, colMeanA, colSqA, scal, out);
}